// SELayer_42271068127525
// MI455X (gfx1250) — compile-verified
//
#include <hip/hip_runtime.h>
#include <math.h>

// Problem constants (from reference)
#define CCH 512             // channels
#define CRR 32              // bottleneck (C/R = 512/16)
#define BB  32              // batch
#define HW  3136            // 56*56
#define NPLANES (BB * CCH)  // 16384 planes of 3136 f32 each

typedef float f32x2 __attribute__((ext_vector_type(2)));
typedef float f32x4 __attribute__((ext_vector_type(4)));
typedef float f32x8 __attribute__((ext_vector_type(8)));

// ---------------------------------------------------------------------------
// CDNA5 async global->LDS DMA (ASYNCcnt-tracked), 16 bytes per lane.
// GV addressing: per-lane 64-bit global address, per-lane 32-bit LDS address.
// Generic-pointer low 32 bits == LDS byte offset (LDS aperture rule).
// ---------------------------------------------------------------------------
__device__ __forceinline__ void async_cp16(const float* lds_dst, const float* gsrc) {
    unsigned lds_off = (unsigned)(size_t)(const void*)lds_dst;
    asm volatile("global_load_async_to_lds_b128 %0, %1, off"
                 :: "v"(lds_off), "v"(gsrc)
                 : "memory");
}
__device__ __forceinline__ void async_wait0() {
    asm volatile("s_wait_asynccnt 0" ::: "memory");
}

// ---------------------------------------------------------------------------
// Kernel 1: global average pool. One block per (b,c) plane (12.5 KB contiguous).
// Default RT temporal hint: we WANT x resident in L2 for the second pass.
// ---------------------------------------------------------------------------
__global__ __launch_bounds__(256) void k_pool(const float* __restrict__ x,
                                              float* __restrict__ y) {
    const int plane = blockIdx.x;
    const f32x4* xp = (const f32x4*)(x + (size_t)plane * HW);
    float s = 0.0f;
    for (int i = threadIdx.x; i < HW / 4; i += 256) {
        f32x4 v = xp[i];
        s += (v.x + v.y) + (v.z + v.w);
    }
    for (int o = 16; o > 0; o >>= 1) s += __shfl_down(s, o, 32);   // wave32
    __shared__ float wsum[8];
    const int wid = threadIdx.x >> 5;
    if ((threadIdx.x & 31) == 0) wsum[wid] = s;
    __syncthreads();
    if (threadIdx.x == 0) {
        float t = 0.0f;
        #pragma unroll
        for (int i = 0; i < 8; ++i) t += wsum[i];
        y[plane] = t * (1.0f / HW);
    }
}

// ---------------------------------------------------------------------------
// Kernel 2: h = relu(y @ w1^T) [32x32,K=512], g = sigmoid(h @ w2^T) [32x512,K=32]
// via V_WMMA_F32_16X16X4_F32. Single block, 8 waves, all operands staged into
// LDS via async DMA. GEMM1 K-split across wave pairs; w1's LDS region is
// re-staged with w2 between the GEMMs. comp = y*(1-g) fused into the epilogue.
//
// LDS layout (floats): sy[16384] | sw[16384] | hp[2048] | hs[1024]  = 140 KB
// (single workgroup; CDNA5 allows up to 320 KB LDS per workgroup)
// ---------------------------------------------------------------------------
__global__ __launch_bounds__(256) void k_gate(const float* __restrict__ y,
                                              const float* __restrict__ w1,
                                              const float* __restrict__ w2,
                                              float* __restrict__ g,
                                              float* __restrict__ comp) {
    extern __shared__ float smem[];
    float* sy = smem;            // y:  32 x 512
    float* sw = smem + 16384;    // w1 (then re-staged with w2)
    float* hp = smem + 32768;    // GEMM1 partial tiles, 8 waves x 256
    float* hs = smem + 34816;    // h: 32 x 32

    const int tid   = threadIdx.x;
    const int lane  = tid & 31;
    const int wid   = tid >> 5;
    const int l15   = lane & 15;
    const int khalf = (lane >> 4) << 1;   // frag K offset: 0 or 2
    const int mofs  = (lane >> 4) << 3;   // C/D row offset: +8 for upper lanes

    // ---- stage y and w1 via async DMA, 16 B per op ----
    for (int i = tid; i < 4096; i += 256) {
        async_cp16(&sy[i * 4], y  + i * 4);
        async_cp16(&sw[i * 4], w1 + i * 4);
    }
    async_wait0();
    __syncthreads();

    // ---- GEMM1: tile t = wid&3 (2x2 over 32x32), K-half = (wid>>2)*256 ----
    {
        const int t  = wid & 3;
        const int kb = (wid >> 2) << 8;
        const int tm = t >> 1, tn = t & 1;
        const float* ya = sy + (tm * 16 + l15) * CCH + kb;
        const float* wb = sw + (tn * 16 + l15) * CCH + kb;
        f32x8 a0 = {}, a1 = {}, a2 = {}, a3 = {};
        for (int k = 0; k < 256; k += 16) {
            f32x2 a, b;
            a = *(const f32x2*)(ya + k + khalf);
            b = *(const f32x2*)(wb + k + khalf);
            a0 = __builtin_amdgcn_wmma_f32_16x16x4_f32(false, a, false, b, (short)0, a0, false, false);
            a = *(const f32x2*)(ya + k + 4 + khalf);
            b = *(const f32x2*)(wb + k + 4 + khalf);
            a1 = __builtin_amdgcn_wmma_f32_16x16x4_f32(false, a, false, b, (short)0, a1, false, false);
            a = *(const f32x2*)(ya + k + 8 + khalf);
            b = *(const f32x2*)(wb + k + 8 + khalf);
            a2 = __builtin_amdgcn_wmma_f32_16x16x4_f32(false, a, false, b, (short)0, a2, false, false);
            a = *(const f32x2*)(ya + k + 12 + khalf);
            b = *(const f32x2*)(wb + k + 12 + khalf);
            a3 = __builtin_amdgcn_wmma_f32_16x16x4_f32(false, a, false, b, (short)0, a3, false, false);
        }
        f32x8 acc = (a0 + a1) + (a2 + a3);
        #pragma unroll
        for (int i = 0; i < 8; ++i)
            hp[wid * 256 + i * 32 + lane] = acc[i];
    }
    __syncthreads();

    // ---- re-stage sw with w2 (async) while combining partials -> hs (relu) ----
    for (int i = tid; i < 4096; i += 256)
        async_cp16(&sw[i * 4], w2 + i * 4);
    for (int e = tid; e < 1024; e += 256) {
        const int t = e >> 8, rem = e & 255, i = rem >> 5, ln = rem & 31;
        const float v = hp[e] + hp[1024 + e];
        const int m = (t >> 1) * 16 + i + ((ln >> 4) << 3);
        const int n = (t & 1) * 16 + (ln & 15);
        hs[m * CRR + n] = v > 0.0f ? v : 0.0f;
    }
    async_wait0();
    __syncthreads();

    // ---- GEMM2: 64 tiles (2 over batch x 32 over channel), 8 per wave, K=32 ----
    for (int t = wid; t < 64; t += 8) {
        const int tm = t & 1;
        const int tn = t >> 1;
        const float* ha = hs + (tm * 16 + l15) * CRR;
        const float* wb = sw + (tn * 16 + l15) * CRR;
        f32x8 a0 = {}, a1 = {};
        #pragma unroll
        for (int k = 0; k < CRR; k += 8) {
            f32x2 a, b;
            a = *(const f32x2*)(ha + k + khalf);
            b = *(const f32x2*)(wb + k + khalf);
            a0 = __builtin_amdgcn_wmma_f32_16x16x4_f32(false, a, false, b, (short)0, a0, false, false);
            a = *(const f32x2*)(ha + k + 4 + khalf);
            b = *(const f32x2*)(wb + k + 4 + khalf);
            a1 = __builtin_amdgcn_wmma_f32_16x16x4_f32(false, a, false, b, (short)0, a1, false, false);
        }
        f32x8 acc = a0 + a1;
        const int mb = tm * 16 + mofs;
        const int nc = tn * 16 + l15;
        #pragma unroll
        for (int i = 0; i < 8; ++i) {
            const int b = mb + i;
            const float s = 1.0f / (1.0f + __expf(-acc[i]));   // sigmoid
            g[b * CCH + nc]    = s;
            comp[b * CCH + nc] = sy[b * CCH + nc] * (1.0f - s); // mean(x*(1-g)) = y*(1-g)
        }
    }
}

// ---------------------------------------------------------------------------
// Kernel 3: scaled = x * g (broadcast per plane). Planes walked in REVERSE so
// the tail of x (still resident in the 192 MB L2 after kernel 1) hits first.
// Output is write-once -> non-temporal stores keep x lines in L2.
// ---------------------------------------------------------------------------
__global__ __launch_bounds__(256) void k_scale(const float* __restrict__ x,
                                               const float* __restrict__ g,
                                               float* __restrict__ out) {
    const int plane = (int)gridDim.x - 1 - (int)blockIdx.x;
    const float gate = g[plane];
    const f32x4* xp = (const f32x4*)(x + (size_t)plane * HW);
    f32x4* op = (f32x4*)(out + (size_t)plane * HW);
    for (int i = threadIdx.x; i < HW / 4; i += 256) {
        f32x4 v = xp[i];
        __builtin_nontemporal_store(v * gate, op + i);
    }
}

extern "C" void kernel_launch(void* const* d_in, const int* in_sizes, int n_in,
                              void* d_out, int out_size, void* d_ws, size_t ws_size,
                              hipStream_t stream) {
    const float* x  = (const float*)d_in[0];   // (32, 512, 56, 56)
    const float* w1 = (const float*)d_in[1];   // (32, 512)
    const float* w2 = (const float*)d_in[2];   // (512, 32)
    float* out    = (float*)d_out;
    float* scaled = out;                                   // (32,512,56,56)
    float* comp   = out + (size_t)BB * CCH * HW;           // (32,512)
    float* y      = (float*)d_ws;                          // 16384 f32
    float* g      = y + BB * CCH;                          // 16384 f32

    const int gate_lds = 35840 * (int)sizeof(float);       // 140 KB dynamic LDS
    (void)hipFuncSetAttribute((const void*)k_gate,
                              hipFuncAttributeMaxDynamicSharedMemorySize, gate_lds);

    k_pool <<<NPLANES, 256, 0,        stream>>>(x, y);
    k_gate <<<1,       256, gate_lds, stream>>>(y, w1, w2, g, comp);
    k_scale<<<NPLANES, 256, 0,        stream>>>(x, g, scaled);
}